// GRDPINet_74053826117824
// MI455X (gfx1250) — compile-verified
//
#include <hip/hip_runtime.h>
#include <cstdint>

// ---------------------------------------------------------------------------
// GRDPINet scan, MI455X (gfx1250).
//
// Math: each step is z <- min(max(z, u-r), u+r) (the reference relu chain is
// algebraically clamp).  Clamp maps f(z)=min(max(z,L),H) compose:
//   (l,h) after (L,H)  ==>  L'=max(L,l), H'=min(max(H,l),h)   (exact in FP)
// => segmented parallel scan over T:
//   K1: per (row, segment) fold 256 steps into (L[4],H[4]) summaries
//   K2: per row, exclusive prefix over 16 summaries (+ z_final, + weights)
//   K3: per (row, segment) start from clamp(z_init, prefix), replay the exact
//       reference arithmetic, emit outputs.
// Bandwidth-bound: ~192 MB HBM traffic => ~8.3 us floor at 23.3 TB/s.
//
// Data movement (CDNA5 paths):
//   inputs : TENSOR_LOAD_TO_LDS (TDM, one 2D strided tile per descriptor,
//            hardware pad to a 36-float LDS row stride), TENSORcnt
//   outputs: GLOBAL_STORE_ASYNC_FROM_LDS_B128 per lane, ASYNCcnt
// ---------------------------------------------------------------------------

#define TT      4096                 // time steps
#define BB      4096                 // batch rows
#define NSEG    16
#define SEGLEN  (TT / NSEG)          // 256
#define ROWS    128                  // rows per block == threads per block
#define CHUNK   32                   // timesteps staged per tile
#define NCHUNK  (SEGLEN / CHUNK)     // 8
#define LSTR    36                   // padded floats per LDS row (16B aligned)
#define F4PT    8                    // float4 store issues per thread per tile

typedef uint32_t u32x4 __attribute__((ext_vector_type(4)));
typedef uint32_t u32x8 __attribute__((ext_vector_type(8)));

__device__ __forceinline__ uint32_t lds_addr32(const void* p) {
  // AS3 -> generic cast: workgroup-relative LDS offset sits in the low 32
  // bits of the flat address (shared aperture lives in bits [63:32]).
  return (uint32_t)(uintptr_t)p;
}

// TDM: load a 2D tile (tile_dim0=CHUNK f32 cols, tile_dim1=ROWS rows, row
// stride TT elements) from global into LDS, padding each 32-DWORD row with
// 4 DWORDs so LDS rows land on a 36-float stride (conflict-friendly layout).
__device__ __forceinline__ void tdm_load_tile(uint32_t lds_byte_addr,
                                              const float* gtile) {
  const uint64_t ga = (uint64_t)(uintptr_t)gtile;
  u32x4 g0;
  g0[0] = 0x1u;                                 // count=1 (valid user D#)
  g0[1] = lds_byte_addr;                        // lds_addr (bytes)
  g0[2] = (uint32_t)ga;                         // global_addr[31:0]
  g0[3] = (uint32_t)((ga >> 32) & 0x1FFFFFFu)   // global_addr[56:32]
          | (2u << 30);                         // type = 2 ("image")
  u32x8 g1;
  g1[0] = (2u << 16)                            // data_size = 4 bytes
        | (1u << 20)                            // pad_enable
        | (4u << 22)                            // pad_interval: 2^(4+1)=32 DW
        | (3u << 25);                           // pad_amount: 3+1 = 4 DWORDs
  g1[1] = ((uint32_t)TT & 0xFFFFu) << 16;       // tensor_dim0[15:0]
  g1[2] = ((uint32_t)TT >> 16)                  // tensor_dim0[31:16]
        | (((uint32_t)TT & 0xFFFFu) << 16);     // tensor_dim1[15:0]
  g1[3] = ((uint32_t)TT >> 16)                  // tensor_dim1[31:16]
        | ((uint32_t)CHUNK << 16);              // tile_dim0 = 32 elements
  g1[4] = (uint32_t)ROWS;                       // tile_dim1 = 128; tile_dim2=0
  g1[5] = (uint32_t)TT;                         // tensor_dim0_stride[31:0]
  g1[6] = 0u;                                   // stride hi / dim1_stride lo
  g1[7] = 0u;                                   // dim1_stride hi (unused, 2D)
  asm volatile("tensor_load_to_lds %0, %1" :: "s"(g0), "s"(g1) : "memory");
}

__device__ __forceinline__ void wait_tensor0() {
  asm volatile("s_wait_tensorcnt 0" ::: "memory");
}
__device__ __forceinline__ void async_st_b128(uint32_t voff, uint32_t lds,
                                              void* sbase) {
  asm volatile("global_store_async_from_lds_b128 %0, %1, %2"
               :: "v"(voff), "v"(lds), "s"(sbase) : "memory");
}
__device__ __forceinline__ void wait_async_le8() {
  asm volatile("s_wait_asynccnt 8" ::: "memory");   // stores retire in order
}

// ---------------------------------------------------------------------------
// K1: segment summaries.  Block = 128 threads = 128 rows; grid (B/128, NSEG).
// ---------------------------------------------------------------------------
__global__ __launch_bounds__(ROWS) void seg_summary_kernel(
    const float* __restrict__ x, const float* __restrict__ xp,
    float* __restrict__ sums)
{
  __shared__ __align__(16) float sx[2][ROWS * LSTR];
  __shared__ __align__(16) float sp[2][ROWS * LSTR];

  const int tid     = threadIdx.x;
  const int rowBase = blockIdx.x * ROWS;
  const int seg     = blockIdx.y;
  const int t0      = seg * SEGLEN;

  auto issue_tiles = [&](int buf, int tcur) {      // wave-0 only, TDM
    const size_t g = (size_t)rowBase * TT + (size_t)tcur;
    tdm_load_tile(lds_addr32(&sx[buf][0]), x + g);
    tdm_load_tile(lds_addr32(&sp[buf][0]), xp + g);
  };

  if (tid < 32) issue_tiles(0, t0);

  float L0 = -__builtin_inff(), L1 = L0, L2 = L0, L3 = L0;
  float H0 =  __builtin_inff(), H1 = H0, H2 = H0, H3 = H0;

  for (int ic = 0; ic < NCHUNK; ++ic) {
    if (tid < 32) wait_tensor0();
    __syncthreads();                               // chunk ic visible to all
    if (tid < 32 && ic + 1 < NCHUNK)
      issue_tiles((ic + 1) & 1, t0 + (ic + 1) * CHUNK);

    const float* rx = &sx[ic & 1][tid * LSTR];
    const float* rp = &sp[ic & 1][tid * LSTR];
#pragma unroll 4
    for (int c = 0; c < CHUNK; ++c) {
      const float u  = rx[c];
      const float up = rp[c];
      const float du = u - up;
      const float s  = 2.0f * sqrtf(du * du + 1e-10f);
      float lo, hi;
      lo = u - (0.1f + s); hi = u + (0.1f + s);
      L0 = fmaxf(L0, lo);  H0 = fminf(fmaxf(H0, lo), hi);
      lo = u - (0.2f + s); hi = u + (0.2f + s);
      L1 = fmaxf(L1, lo);  H1 = fminf(fmaxf(H1, lo), hi);
      lo = u - (0.3f + s); hi = u + (0.3f + s);
      L2 = fmaxf(L2, lo);  H2 = fminf(fmaxf(H2, lo), hi);
      lo = u - (0.4f + s); hi = u + (0.4f + s);
      L3 = fmaxf(L3, lo);  H3 = fminf(fmaxf(H3, lo), hi);
    }
    __syncthreads();                               // done reading buf[ic&1]
  }

  const size_t base = ((size_t)seg * BB + (size_t)(rowBase + tid)) * 8;
  float4* o = (float4*)(sums + base);
  o[0] = make_float4(L0, L1, L2, L3);
  o[1] = make_float4(H0, H1, H2, H3);
}

// ---------------------------------------------------------------------------
// K2: per-row exclusive prefix over segment summaries; z_final; weights.
// ---------------------------------------------------------------------------
__global__ void prefix_kernel(const float* __restrict__ sums,
                              float* __restrict__ pre,
                              const float* __restrict__ zinit,
                              const float* __restrict__ w,
                              float* __restrict__ out)
{
  const int r = blockIdx.x * blockDim.x + threadIdx.x;
  if (r < 11) out[(size_t)BB * TT + r] = w[r];     // weights passthrough
  if (r >= BB) return;

  float L0 = -__builtin_inff(), L1 = L0, L2 = L0, L3 = L0;
  float H0 =  __builtin_inff(), H1 = H0, H2 = H0, H3 = H0;

  for (int s = 0; s < NSEG; ++s) {
    const size_t idx = ((size_t)s * BB + (size_t)r) * 8;
    float4* p = (float4*)(pre + idx);
    p[0] = make_float4(L0, L1, L2, L3);            // exclusive prefix
    p[1] = make_float4(H0, H1, H2, H3);
    const float4 lv = ((const float4*)(sums + idx))[0];
    const float4 hv = ((const float4*)(sums + idx))[1];
    L0 = fmaxf(L0, lv.x); H0 = fminf(fmaxf(H0, lv.x), hv.x);
    L1 = fmaxf(L1, lv.y); H1 = fminf(fmaxf(H1, lv.y), hv.y);
    L2 = fmaxf(L2, lv.z); H2 = fminf(fmaxf(H2, lv.z), hv.z);
    L3 = fmaxf(L3, lv.w); H3 = fminf(fmaxf(H3, lv.w), hv.w);
  }

  const float4 z0 = ((const float4*)zinit)[r];
  float* zf = out + (size_t)BB * TT + 11 + (size_t)r * 4;   // 4B-aligned only
  zf[0] = fminf(fmaxf(z0.x, L0), H0);
  zf[1] = fminf(fmaxf(z0.y, L1), H1);
  zf[2] = fminf(fmaxf(z0.z, L2), H2);
  zf[3] = fminf(fmaxf(z0.w, L3), H3);
}

// ---------------------------------------------------------------------------
// K3: apply pass.  TDM tile loads; exact reference replay; outputs staged in
// LDS and written with per-lane async stores-from-LDS (coalesced mapping).
// ---------------------------------------------------------------------------
__global__ __launch_bounds__(ROWS) void apply_kernel(
    const float* __restrict__ x, const float* __restrict__ xp,
    const float* __restrict__ pre, const float* __restrict__ zinit,
    float* __restrict__ out)
{
  __shared__ __align__(16) float sx[2][ROWS * LSTR];
  __shared__ __align__(16) float sp[2][ROWS * LSTR];
  __shared__ __align__(16) float so[2][ROWS * LSTR];

  const int tid     = threadIdx.x;
  const int rowBase = blockIdx.x * ROWS;
  const int seg     = blockIdx.y;
  const int t0      = seg * SEGLEN;
  const int row     = rowBase + tid;

  auto issue_tiles = [&](int buf, int tcur) {      // wave-0 only, TDM
    const size_t g = (size_t)rowBase * TT + (size_t)tcur;
    tdm_load_tile(lds_addr32(&sx[buf][0]), x + g);
    tdm_load_tile(lds_addr32(&sp[buf][0]), xp + g);
  };

  if (tid < 32) issue_tiles(0, t0);

  // z at segment start = clamp(z_init, exclusive-prefix composition)
  const size_t pidx = ((size_t)seg * BB + (size_t)row) * 8;
  const float4 PL = ((const float4*)(pre + pidx))[0];
  const float4 PH = ((const float4*)(pre + pidx))[1];
  const float4 zi = ((const float4*)zinit)[row];
  float z0v = fminf(fmaxf(zi.x, PL.x), PH.x);
  float z1v = fminf(fmaxf(zi.y, PL.y), PH.y);
  float z2v = fminf(fmaxf(zi.z, PL.z), PH.z);
  float z3v = fminf(fmaxf(zi.w, PL.w), PH.w);

  for (int ic = 0; ic < NCHUNK; ++ic) {
    wait_async_le8();          // drain my stores older than last chunk's 8
    if (tid < 32) wait_tensor0();
    __syncthreads();           // chunk ic inputs visible; so[ic&1] reusable
    if (tid < 32 && ic + 1 < NCHUNK)
      issue_tiles((ic + 1) & 1, t0 + (ic + 1) * CHUNK);

    const float* rx = &sx[ic & 1][tid * LSTR];
    const float* rp = &sp[ic & 1][tid * LSTR];
    float*       ro = &so[ic & 1][tid * LSTR];
#pragma unroll 4
    for (int c = 0; c < CHUNK; ++c) {
      const float u  = rx[c];
      const float up = rp[c];
      const float du = u - up;
      const float s  = 2.0f * sqrtf(du * du + 1e-10f);
      float rr, lo, hi;
      // exact reference chain: z = lo + relu(hi - relu(hi - z) - lo)
      rr = 0.1f + s; lo = u - rr; hi = u + rr;
      z0v = lo + fmaxf(hi - fmaxf(hi - z0v, 0.0f) - lo, 0.0f);
      rr = 0.2f + s; lo = u - rr; hi = u + rr;
      z1v = lo + fmaxf(hi - fmaxf(hi - z1v, 0.0f) - lo, 0.0f);
      rr = 0.3f + s; lo = u - rr; hi = u + rr;
      z2v = lo + fmaxf(hi - fmaxf(hi - z2v, 0.0f) - lo, 0.0f);
      rr = 0.4f + s; lo = u - rr; hi = u + rr;
      z3v = lo + fmaxf(hi - fmaxf(hi - z3v, 0.0f) - lo, 0.0f);
      ro[c] = 0.1f * u + 0.2f * z0v + 0.3f * z1v + 0.4f * z2v + 0.5f * z3v;
    }
    __syncthreads();           // output tile fully staged in LDS

    {  // coalesced async stores LDS -> out (8 lanes cover 128B per row)
      const uint32_t lo32 = lds_addr32(&so[ic & 1][0]);
#pragma unroll
      for (int j = 0; j < F4PT; ++j) {
        const int idx = j * ROWS + tid;
        const int r   = idx >> 3;
        const int q   = idx & 7;
        const uint32_t goff =
            (uint32_t)((((rowBase + r) * TT) + t0 + ic * CHUNK + q * 4) * 4);
        const uint32_t loff = (uint32_t)((r * LSTR + q * 4) * 4);
        async_st_b128(goff, lo32 + loff, out);
      }
    }
  }
  // s_endpgm performs an implicit wait-idle, draining ASYNCcnt/TENSORcnt.
}

// ---------------------------------------------------------------------------
extern "C" void kernel_launch(void* const* d_in, const int* in_sizes, int n_in,
                              void* d_out, int out_size, void* d_ws,
                              size_t ws_size, hipStream_t stream)
{
  const float* x  = (const float*)d_in[0];   // (B, T, 1)
  const float* xp = (const float*)d_in[1];   // (B, T, 1)
  const float* z0 = (const float*)d_in[2];   // (B, 1, 4)
  const float* w  = (const float*)d_in[3];   // (11,)
  float* out = (float*)d_out;                // out(B*T) | weights(11) | z_final(B*4)

  float* sums = (float*)d_ws;                         // NSEG*B*8 floats = 2 MB
  float* pre  = sums + (size_t)NSEG * BB * 8;         // NSEG*B*8 floats = 2 MB

  dim3 tileGrid(BB / ROWS, NSEG);                     // (32, 16)
  seg_summary_kernel<<<tileGrid, ROWS, 0, stream>>>(x, xp, sums);
  prefix_kernel<<<BB / 256, 256, 0, stream>>>(sums, pre, z0, w, out);
  apply_kernel<<<tileGrid, ROWS, 0, stream>>>(x, xp, pre, z0, out);
}